// GINNet_9251359555639
// MI455X (gfx1250) — compile-verified
//
#include <hip/hip_runtime.h>

typedef float v2f __attribute__((ext_vector_type(2)));
typedef float v8f __attribute__((ext_vector_type(8)));

#define D_FEAT 128
#define H_FEAT 256
#define NGRAPH 256
#define BN_EPS 1e-5f

// ---------------- zero fill ----------------
__global__ void zero_f32(float* __restrict__ p, int n) {
  int i = blockIdx.x * blockDim.x + threadIdx.x;
  if (i < n) p[i] = 0.f;
}

// ---------------- scatter-add: agg[dst] += h[src]  (one wave per edge) ----------------
__global__ void scatter_add(const float* __restrict__ h,
                            const int* __restrict__ src,
                            const int* __restrict__ dst,
                            float* __restrict__ agg, int E) {
  int t = blockIdx.x * blockDim.x + threadIdx.x;
  int e = t >> 5;
  if (e >= E) return;
  int lane = t & 31;                      // 32 lanes * float4 = 128 features
  int s = src[e], d = dst[e];
  float4 v = ((const float4*)(h + (size_t)s * D_FEAT))[lane];
  float* b = agg + (size_t)d * D_FEAT + lane * 4;
  atomicAdd(b + 0, v.x);
  atomicAdd(b + 1, v.y);
  atomicAdd(b + 2, v.z);
  atomicAdd(b + 3, v.w);
}

// ---------------- f32 WMMA GEMM with fused epilogues ----------------
//   Y[N,NC] = Apre @ W[K,NC] + bias, accumulating per-column sum/sumsq (BN stats)
//   PRE_BN:  Apre = relu(A * pre_sc[k] + pre_sh[k])   (fused BN1+ReLU on the fly)
// 256 threads = 8 wave32s in a 2x4 grid; block tile = 32 rows x 64 cols.
template <bool PRE_BN>
__global__ __launch_bounds__(256) void gemm_wmma_f32(
    const float* __restrict__ A, const float* __restrict__ W,
    const float* __restrict__ bias,
    const float* __restrict__ pre_sc, const float* __restrict__ pre_sh,
    float* __restrict__ Y, float* __restrict__ osum, float* __restrict__ osq,
    int N, int K, int NC) {
  __shared__ float s_red[128];            // [0..63] col sums, [64..127] col sumsq
  if (threadIdx.x < 128) s_red[threadIdx.x] = 0.f;
  __syncthreads();

  int lane  = threadIdx.x & 31;
  int wave  = threadIdx.x >> 5;
  int waveM = wave >> 2;                  // 0..1
  int waveN = wave & 3;                   // 0..3
  int rowBase = blockIdx.x * 32 + waveM * 16;
  int colBase = blockIdx.y * 64 + waveN * 16;
  int m    = lane & 15;
  int half = lane >> 4;                   // lane half selects K pair {0,1} vs {2,3}
  int col  = colBase + m;

  int arow = rowBase + m;                 // clamped load row; stores/stats masked
  if (arow >= N) arow = N - 1;
  const float* ap  = A + (size_t)arow * K + 2 * half;
  const float* wp  = W + (size_t)(2 * half) * NC + col;
  const float* psc = PRE_BN ? (pre_sc + 2 * half) : nullptr;
  const float* psh = PRE_BN ? (pre_sh + 2 * half) : nullptr;

  v8f acc = {};
  for (int k = 0; k < K; k += 4) {
    float a0 = ap[0];                     // A[m][k + 2*half]
    float a1 = ap[1];                     // A[m][k + 2*half + 1]
    if (PRE_BN) {                         // fused BN + ReLU on A operand
      a0 = fmaxf(a0 * psc[k] + psh[k], 0.f);
      a1 = fmaxf(a1 * psc[k + 1] + psh[k + 1], 0.f);
    }
    v2f a, b;
    a.x = a0; a.y = a1;
    b.x = wp[0];                          // W[k + 2*half][col]
    b.y = wp[NC];                         // W[k + 2*half + 1][col]
    acc = __builtin_amdgcn_wmma_f32_16x16x4_f32(
        /*neg_a=*/false, a, /*neg_b=*/false, b,
        /*c_mod=*/(short)0, acc, /*reuse_a=*/false, /*reuse_b=*/false);
    ap += 4;
    wp += (size_t)4 * NC;
  }

  float bv = bias[col];
  float lsum = 0.f, lsq = 0.f;
#pragma unroll
  for (int i = 0; i < 8; ++i) {           // C layout: vgpr i -> rows i / i+8 by lane half
    int row = rowBase + i + half * 8;
    float v = acc[i] + bv;
    if (row < N) {
      Y[(size_t)row * NC + col] = v;
      lsum += v;
      lsq  += v * v;
    }
  }
  int colLocal = waveN * 16 + m;          // 0..63 within block tile
  atomicAdd(&s_red[colLocal], lsum);      // LDS ds_add_f32 reduction
  atomicAdd(&s_red[64 + colLocal], lsq);
  __syncthreads();
  if (threadIdx.x < 64)
    atomicAdd(&osum[blockIdx.y * 64 + threadIdx.x], s_red[threadIdx.x]);
  else if (threadIdx.x < 128)
    atomicAdd(&osq[blockIdx.y * 64 + threadIdx.x - 64], s_red[threadIdx.x]);
}

// ---------------- BN scale/shift from stats ----------------
__global__ void bn_finalize(const float* __restrict__ sum, const float* __restrict__ sumsq,
                            const float* __restrict__ g, const float* __restrict__ bt,
                            float* __restrict__ scale, float* __restrict__ shift,
                            int C, float invN) {
  int c = blockIdx.x * blockDim.x + threadIdx.x;
  if (c >= C) return;
  float mn = sum[c] * invN;
  float vr = sumsq[c] * invN - mn * mn;   // biased variance (matches jnp.var)
  float s  = g[c] * rsqrtf(vr + BN_EPS);
  scale[c] = s;
  shift[c] = bt[c] - mn * s;
}

// ---------------- apply BN (+optional ReLU), float4 vectorized ----------------
__global__ void bn_apply(const float* __restrict__ Y, const float* __restrict__ scale,
                         const float* __restrict__ shift, float* __restrict__ out,
                         int total4, int C, int do_relu) {
  int i = blockIdx.x * blockDim.x + threadIdx.x;
  if (i >= total4) return;
  int c4 = (i % (C >> 2)) << 2;           // column of first element (C % 4 == 0)
  float4 v = ((const float4*)Y)[i];
  v.x = v.x * scale[c4 + 0] + shift[c4 + 0];
  v.y = v.y * scale[c4 + 1] + shift[c4 + 1];
  v.z = v.z * scale[c4 + 2] + shift[c4 + 2];
  v.w = v.w * scale[c4 + 3] + shift[c4 + 3];
  if (do_relu) {
    v.x = fmaxf(v.x, 0.f); v.y = fmaxf(v.y, 0.f);
    v.z = fmaxf(v.z, 0.f); v.w = fmaxf(v.w, 0.f);
  }
  ((float4*)out)[i] = v;
}

// ---------------- per-graph pooling accumulate ----------------
__global__ void pool_accum(const float* __restrict__ h, const int* __restrict__ batch,
                           float* __restrict__ pools, float* __restrict__ cnts, int N) {
  int t = blockIdx.x * blockDim.x + threadIdx.x;
  int node = t >> 5;
  if (node >= N) return;
  int lane = t & 31;
  int g = batch[node];
  float4 v = ((const float4*)(h + (size_t)node * D_FEAT))[lane];
  float* b = pools + (size_t)g * D_FEAT + lane * 4;
  atomicAdd(b + 0, v.x);
  atomicAdd(b + 1, v.y);
  atomicAdd(b + 2, v.z);
  atomicAdd(b + 3, v.w);
  if (lane == 0) atomicAdd(&cnts[g], 1.f);
}

// ---------------- classifier: out[g,c] = mean_pool[g] . Wc[:,c] + bc[c] ----------------
__global__ void classify(const float* __restrict__ pools, const float* __restrict__ cnts,
                         const float* __restrict__ Wc, const float* __restrict__ bc,
                         float* __restrict__ out, int C) {
  int t = blockIdx.x * blockDim.x + threadIdx.x;
  if (t >= NGRAPH * C) return;
  int g = t / C, c = t % C;
  float inv = 1.f / fmaxf(cnts[g], 1.f);
  const float* p = pools + (size_t)g * D_FEAT;
  float s = 0.f;
  for (int d = 0; d < D_FEAT; ++d) s += p[d] * Wc[d * C + c];
  out[t] = s * inv + bc[c];
}

extern "C" void kernel_launch(void* const* d_in, const int* in_sizes, int n_in,
                              void* d_out, int out_size, void* d_ws, size_t ws_size,
                              hipStream_t stream) {
  const float* x     = (const float*)d_in[0];
  const int*   ei    = (const int*)  d_in[1];
  const int*   batch = (const int*)  d_in[2];
  const float* W1    = (const float*)d_in[3];
  const float* b1    = (const float*)d_in[4];
  const float* g1    = (const float*)d_in[5];
  const float* bt1   = (const float*)d_in[6];
  const float* W2    = (const float*)d_in[7];
  const float* b2    = (const float*)d_in[8];
  const float* g2    = (const float*)d_in[9];
  const float* bt2   = (const float*)d_in[10];
  const float* Wc    = (const float*)d_in[11];
  const float* bc    = (const float*)d_in[12];

  const int N = in_sizes[0] / D_FEAT;
  const int E = in_sizes[1] / 2;
  const int L = in_sizes[3] / (D_FEAT * H_FEAT);
  const int C = in_sizes[12];             // classes
  const int* src = ei;
  const int* dst = ei + E;

  // workspace layout
  char* ws = (char*)d_ws;
  size_t off = 0;
  auto alloc = [&](size_t bytes) -> float* {
    float* p = (float*)(ws + off);
    off += (bytes + 255) & ~(size_t)255;
    return p;
  };
  float* hbuf  = alloc((size_t)N * D_FEAT * sizeof(float));   // h after each layer
  float* aggb  = alloc((size_t)N * D_FEAT * sizeof(float));   // h+agg, then Y2
  float* Y1    = alloc((size_t)N * H_FEAT * sizeof(float));
  float* ssum  = alloc(H_FEAT * sizeof(float));               // ssum/ssq contiguous
  float* ssq   = alloc(H_FEAT * sizeof(float));
  float* sc    = alloc(H_FEAT * sizeof(float));
  float* sh    = alloc(H_FEAT * sizeof(float));
  float* pools = alloc((size_t)NGRAPH * D_FEAT * sizeof(float)); // pools/cnts contiguous
  float* cnts  = alloc(NGRAPH * sizeof(float));

  const int edgeThreads = E * 32;
  const int nodeThreads = N * 32;

  for (int l = 0; l < L; ++l) {
    const float* h = (l == 0) ? x : hbuf;

    // agg = h ; agg += scatter(h[src] -> dst)
    hipMemcpyAsync(aggb, h, (size_t)N * D_FEAT * sizeof(float),
                   hipMemcpyDeviceToDevice, stream);
    scatter_add<<<(edgeThreads + 255) / 256, 256, 0, stream>>>(h, src, dst, aggb, E);

    // Y1 = (h+agg) @ W1[l] + b1[l]   (+ fused BN1 column stats)
    zero_f32<<<2, 256, 0, stream>>>(ssum, 512);   // zeros ssum+ssq
    {
      dim3 grid((N + 31) / 32, H_FEAT / 64);
      gemm_wmma_f32<false><<<grid, 256, 0, stream>>>(
          aggb, W1 + (size_t)l * D_FEAT * H_FEAT, b1 + (size_t)l * H_FEAT,
          nullptr, nullptr, Y1, ssum, ssq, N, D_FEAT, H_FEAT);
    }
    bn_finalize<<<1, H_FEAT, 0, stream>>>(ssum, ssq, g1 + (size_t)l * H_FEAT,
                                          bt1 + (size_t)l * H_FEAT, sc, sh,
                                          H_FEAT, 1.f / (float)N);

    // Y2 = relu(BN1(Y1)) @ W2[l] + b2[l]   (BN1+ReLU fused into A loads,
    //                                        fused BN2 column stats; into aggb)
    zero_f32<<<2, 256, 0, stream>>>(ssum, 512);
    {
      dim3 grid((N + 31) / 32, D_FEAT / 64);
      gemm_wmma_f32<true><<<grid, 256, 0, stream>>>(
          Y1, W2 + (size_t)l * H_FEAT * D_FEAT, b2 + (size_t)l * D_FEAT,
          sc, sh, aggb, ssum, ssq, N, H_FEAT, D_FEAT);
    }
    bn_finalize<<<1, D_FEAT, 0, stream>>>(ssum, ssq, g2 + (size_t)l * D_FEAT,
                                          bt2 + (size_t)l * D_FEAT, sc, sh,
                                          D_FEAT, 1.f / (float)N);
    // BN2 (+ReLU except last layer) -> hbuf
    bn_apply<<<((N * D_FEAT / 4) + 255) / 256, 256, 0, stream>>>(
        aggb, sc, sh, hbuf, N * D_FEAT / 4, D_FEAT, (l < L - 1) ? 1 : 0);
  }

  // global mean pool + classifier
  zero_f32<<<(NGRAPH * D_FEAT + NGRAPH + 255) / 256, 256, 0, stream>>>(
      pools, NGRAPH * D_FEAT + NGRAPH);
  pool_accum<<<(nodeThreads + 255) / 256, 256, 0, stream>>>(hbuf, batch, pools, cnts, N);
  classify<<<(NGRAPH * C + 255) / 256, 256, 0, stream>>>(pools, cnts, Wc, bc,
                                                         (float*)d_out, C);
}